// SparseToDense_59777354826220
// MI455X (gfx1250) — compile-verified
//
#include <hip/hip_runtime.h>

// SparseToDense scatter-add for MI455X (gfx1250, wave32).
//
// Phase 1: zero the 512MB dense output by streaming a 4KB LDS block of zeros
//          through GLOBAL_STORE_ASYNC_FROM_LDS_B128 (ASYNCcnt-tracked async
//          engine). Block-contiguous tiling: all 16 stores per thread share one
//          VGPR byte-offset and differ only in the 24-bit IOFFSET literal, and
//          fast/slow path selection is block-uniform (scalar branch, no EXEC
//          masking on the hot path).
// Phase 2: one lane per (point, feature): native L2-side GLOBAL_ATOMIC_ADD_F32
//          in SADDR form (uniform base in SGPRs + 32-bit byte offset in VGPR).

constexpr int kFeat      = 16;   // features per point
constexpr int kGridShift = 7;    // 128 = 1<<7 per spatial dim
constexpr int kFillBlockThreads    = 256;
constexpr int kFillChunksPerThread = 16;  // 16B chunks per thread
constexpr unsigned kFillChunksPerBlock =
    kFillBlockThreads * kFillChunksPerThread;  // 4096 chunks = 64KB per block

typedef __attribute__((address_space(3))) const unsigned char lds_byte_t;

__device__ __forceinline__ unsigned lds_offset_of(const void* p) {
  // generic -> LDS(AS3) addrspace cast; AS3 pointers are the raw 32-bit LDS
  // byte offset the async-store instruction expects in its VSRC operand.
  return (unsigned)(unsigned long long)(lds_byte_t*)p;
}

#define S2D_ASYNC_STORE16(OFF)                                              \
  asm volatile("global_store_async_from_lds_b128 %0, %1, %2 offset:" #OFF  \
               :                                                            \
               : "v"(voff), "v"(lds_off), "s"(out)                          \
               : "memory")

__global__ __launch_bounds__(kFillBlockThreads) void s2d_zero_async_kernel(
    float* __restrict__ out, unsigned n16 /* number of 16B chunks */) {
  __shared__ __align__(16) uint4 zbuf[kFillBlockThreads];  // 4KB of zeros

  const unsigned tid = threadIdx.x;
  zbuf[tid] = uint4{0u, 0u, 0u, 0u};   // ds_store_b128
  __syncthreads();                     // DScnt drained before async engine reads

  const unsigned lds_off    = lds_offset_of(&zbuf[tid]);
  const unsigned block_base = blockIdx.x * kFillChunksPerBlock;

  // Block-uniform path selection: scalar s_cmp/s_cbranch, no EXEC masking.
  if (block_base + kFillChunksPerBlock <= n16) {
    // Fast path (every block when n16 is a multiple of 4096): 16 async
    // stores, one shared VGPR offset, literal IOFFSETs. Lanes 0..31 write
    // consecutive 16B chunks -> 512B contiguous per wave-issue.
    const unsigned voff = (block_base + tid) * 16u;   // byte offset, < 2^30
    S2D_ASYNC_STORE16(0);
    S2D_ASYNC_STORE16(4096);
    S2D_ASYNC_STORE16(8192);
    S2D_ASYNC_STORE16(12288);
    S2D_ASYNC_STORE16(16384);
    S2D_ASYNC_STORE16(20480);
    S2D_ASYNC_STORE16(24576);
    S2D_ASYNC_STORE16(28672);
    S2D_ASYNC_STORE16(32768);
    S2D_ASYNC_STORE16(36864);
    S2D_ASYNC_STORE16(40960);
    S2D_ASYNC_STORE16(45056);
    S2D_ASYNC_STORE16(49152);
    S2D_ASYNC_STORE16(53248);
    S2D_ASYNC_STORE16(57344);
    S2D_ASYNC_STORE16(61440);
  } else {
    // Remainder path (cold; unused for this problem size).
    for (int j = 0; j < kFillChunksPerThread; ++j) {
      const unsigned chunk = block_base + tid + (unsigned)j * kFillBlockThreads;
      if (chunk < n16) {
        const unsigned voff = chunk * 16u;
        asm volatile("global_store_async_from_lds_b128 %0, %1, %2"
                     :
                     : "v"(voff), "v"(lds_off), "s"(out)
                     : "memory");
      }
    }
  }
  asm volatile("s_wait_asynccnt 0x0" ::: "memory");
}

__global__ __launch_bounds__(256) void s2d_scatter_kernel(
    const int4* __restrict__ coords,   // [N] of (b, x, y, z)
    const float* __restrict__ feats,   // [N, 16]
    float* __restrict__ out,           // [4, 16, 128, 128, 128]
    int npoints) {
  int gid = blockIdx.x * blockDim.x + threadIdx.x;
  int p = gid >> 4;          // point index (16 lanes per point)
  if (p >= npoints) return;
  int f = gid & (kFeat - 1); // feature channel for this lane

  int4 c = coords[p];        // global_load_b128, broadcast within half-wave
  unsigned x = (unsigned)c.y;
  unsigned y = (unsigned)c.z;
  unsigned z = (unsigned)c.w;
  // Any coordinate <0 or >=128 -> masked out (reference adds +0 there: no-op).
  if ((x | y | z) >= 128u) return;

  // Perfectly coalesced per-wave read of feats (128B / wave).
  float v = feats[(size_t)p * kFeat + f];

  // out[b][f][x][y][z], channel stride = 128^3 = 2^21 elements.
  unsigned bc   = (unsigned)c.x * kFeat + (unsigned)f;
  unsigned boff = ((bc << (3 * kGridShift)) | (x << (2 * kGridShift)) |
                   (y << kGridShift) | z) << 2;   // byte offset, < 2^29

  // Non-returning hardware f32 atomic add (L2 atomic units), SADDR form:
  // uniform 64-bit base in SGPRs, per-lane 32-bit byte offset in a VGPR.
  asm volatile("global_atomic_add_f32 %0, %1, %2"
               :
               : "v"(boff), "v"(v), "s"(out)
               : "memory");
}

extern "C" void kernel_launch(void* const* d_in, const int* in_sizes, int n_in,
                              void* d_out, int out_size, void* d_ws, size_t ws_size,
                              hipStream_t stream) {
  const int4*  coords = (const int4*)d_in[0];  // int32 [N,4]
  const float* feats  = (const float*)d_in[1]; // float32 [N,16]
  float*       out    = (float*)d_out;         // float32 [4,16,128,128,128]

  const int npoints = in_sizes[0] / 4;

  // Phase 1: zero output (must run every launch; harness replays the graph).
  const unsigned n16 = (unsigned)(out_size / 4);  // 16B chunks
  const int fill_blocks =
      (int)((n16 + kFillChunksPerBlock - 1) / kFillChunksPerBlock);
  s2d_zero_async_kernel<<<fill_blocks, kFillBlockThreads, 0, stream>>>(out, n16);

  // Phase 2: scatter-add, one lane per (point, feature).
  const long long total = (long long)npoints * kFeat;
  const int scat_blocks = (int)((total + 255) / 256);
  s2d_scatter_kernel<<<scat_blocks, 256, 0, stream>>>(coords, feats, out, npoints);
}